// TCRSSolver_46265387712741
// MI455X (gfx1250) — compile-verified
//
#include <hip/hip_runtime.h>
#include <math.h>

typedef __attribute__((ext_vector_type(2))) float v2f;
typedef __attribute__((ext_vector_type(8))) float v8f;

#define BATCH   32
#define DMODEL  2048
#define DLAT    512
#define DV      2047          // d_model - 1
#define HALF_DT 0.005f        // DT/2

// ---------------------------------------------------------------------------
// Kernel 1: vs[b, j] = HALF_DT * tanh( sum_l z0[b,l] * W[j,l] + bias[j] )
// One wave per 16x16 output tile; K marched in steps of 4 via
// V_WMMA_F32_16X16X4_F32 (the fp32 WMMA shape on CDNA5).
//
// A-frag (16x4 f32, 2 VGPRs): lane&15 = M row of z0, lane-half picks K {0,1}
// vs {2,3}  -> one float2 load per step.
// B-frag (4x16 f32, 2 VGPRs): lane&15 = N (row j of W), lane-half picks K
// {0,1} vs {2,3} -> one float2 load per step (W row-major over K).
// C/D (16x16 f32, 8 VGPRs): M = reg + 8*(lane>=16), N = lane&15.
// ---------------------------------------------------------------------------
__global__ __launch_bounds__(32) void gemm_tanh_wmma_kernel(
    const float* __restrict__ z0,    // (32, 512)
    const float* __restrict__ W,     // (2047, 512)
    const float* __restrict__ bias,  // (2047,)
    float* __restrict__ vs)          // (32, 2047) out = HALF_DT*tanh(...)
{
    const int lane   = threadIdx.x;          // 0..31
    const int l15    = lane & 15;
    const int laneHi = lane >> 4;            // 0 or 1
    const int koff   = laneHi * 2;           // K sub-offset {0,2}

    const int n0 = blockIdx.x * 16;          // output column tile (j)
    const int m0 = blockIdx.y * 16;          // output row tile (batch)

    // A rows are always valid (BATCH=32). Clamp the single invalid W row
    // (j == 2047 in the last tile); its column is masked on store.
    const int mA = m0 + l15;
    int nB = n0 + l15;
    if (nB >= DV) nB = DV - 1;

    const float* __restrict__ zrow = z0 + mA * DLAT + koff;
    const float* __restrict__ wrow = W  + nB * DLAT + koff;

    v8f c = {};   // f32 accumulator, 8 VGPRs

    #pragma unroll 4
    for (int k = 0; k < DLAT; k += 4) {
        v2f a = *(const v2f*)(zrow + k);     // K = k+koff, k+koff+1
        v2f b = *(const v2f*)(wrow + k);
        // 8 args: (neg_a, A, neg_b, B, c_mod, C, reuse_a, reuse_b)
        c = __builtin_amdgcn_wmma_f32_16x16x4_f32(
                false, a, false, b, (short)0, c, false, false);
    }

    // Epilogue: bias + tanh + scale, store with column guard.
    const int j = n0 + l15;
    if (j < DV) {
        const float bj = bias[j];
        #pragma unroll
        for (int r = 0; r < 8; ++r) {
            const int m = m0 + r + 8 * laneHi;
            vs[m * DV + j] = HALF_DT * tanhf(c[r] + bj);
        }
    }
}

// ---------------------------------------------------------------------------
// Kernel 2: per batch row, build the tridiagonal system
//   a_i = vs[i-1], b_i = 1, c_i = -vs[i]
//   d_i = h[i] + vs[i]*h[i+1] - vs[i-1]*h[i-1]
// and solve with parallel cyclic reduction entirely in LDS.
// One 1024-thread block (32 waves) per batch row; each thread owns 2 rows.
// 11 PCR steps (log2(2048)) instead of a 4096-step serial Thomas chain.
// ---------------------------------------------------------------------------
__global__ __launch_bounds__(1024) void pcr_solve_kernel(
    const float* __restrict__ h,    // (32, 2048)
    const float* __restrict__ vs,   // (32, 2047)
    float* __restrict__ x)          // (32, 2048)
{
    __shared__ float sa[DMODEL];
    __shared__ float sb[DMODEL];
    __shared__ float sc[DMODEL];
    __shared__ float sd[DMODEL];

    const int brow = blockIdx.x;
    const int t    = threadIdx.x;
    const float* __restrict__ hr = h  + brow * DMODEL;
    const float* __restrict__ vr = vs + brow * DV;

    #pragma unroll
    for (int e = 0; e < 2; ++e) {
        const int i = t + e * 1024;
        const float vi   = (i < DV) ? vr[i]     : 0.0f;
        const float vim1 = (i > 0)  ? vr[i - 1] : 0.0f;
        const float hi   = hr[i];
        const float hip  = (i < DV) ? hr[i + 1] : 0.0f;
        const float him  = (i > 0)  ? hr[i - 1] : 0.0f;
        sa[i] = vim1;                      // A[i, i-1] =  dt/2 * v[i-1]
        sb[i] = 1.0f;
        sc[i] = -vi;                       // A[i, i+1] = -dt/2 * v[i]
        sd[i] = hi + vi * hip - vim1 * him;
    }
    __syncthreads();

    for (int s = 1; s < DMODEL; s <<= 1) {
        float na[2], nb[2], nc[2], nd[2];
        #pragma unroll
        for (int e = 0; e < 2; ++e) {
            const int i  = t + e * 1024;
            const float ai = sa[i], bi = sb[i], ci = sc[i], di = sd[i];
            const int il = i - s, ir = i + s;
            float al = 0.0f, bl = 1.0f, cl = 0.0f, dl = 0.0f;
            if (il >= 0)     { al = sa[il]; bl = sb[il]; cl = sc[il]; dl = sd[il]; }
            float ar = 0.0f, br = 1.0f, cr = 0.0f, dr = 0.0f;
            if (ir < DMODEL) { ar = sa[ir]; br = sb[ir]; cr = sc[ir]; dr = sd[ir]; }
            const float k1 = ai / bl;
            const float k2 = ci / br;
            na[e] = -k1 * al;
            nc[e] = -k2 * cr;
            nb[e] = bi - k1 * cl - k2 * ar;
            nd[e] = di - k1 * dl - k2 * dr;
        }
        __syncthreads();
        #pragma unroll
        for (int e = 0; e < 2; ++e) {
            const int i = t + e * 1024;
            sa[i] = na[e]; sb[i] = nb[e]; sc[i] = nc[e]; sd[i] = nd[e];
        }
        __syncthreads();
    }

    #pragma unroll
    for (int e = 0; e < 2; ++e) {
        const int i = t + e * 1024;
        x[brow * DMODEL + i] = sd[i] / sb[i];
    }
}

// ---------------------------------------------------------------------------
extern "C" void kernel_launch(void* const* d_in, const int* in_sizes, int n_in,
                              void* d_out, int out_size, void* d_ws, size_t ws_size,
                              hipStream_t stream) {
    const float* z0   = (const float*)d_in[0];  // (32, 512)
    const float* h    = (const float*)d_in[1];  // (32, 2048)
    const float* W    = (const float*)d_in[2];  // (2047, 512)
    const float* bias = (const float*)d_in[3];  // (2047,)
    float* out = (float*)d_out;                 // (32, 2048)
    float* vs  = (float*)d_ws;                  // (32, 2047) scratch, 262 KB

    // Tile grid: 128 column tiles x 2 batch tiles, one wave each.
    gemm_tanh_wmma_kernel<<<dim3(128, 2), 32, 0, stream>>>(z0, W, bias, vs);
    // One block per batch row; PCR solve in LDS.
    pcr_solve_kernel<<<BATCH, 1024, 0, stream>>>(h, vs, out);
}